// RegionProposalNetwork_26087631356446
// MI455X (gfx1250) — compile-verified
//
#include <hip/hip_runtime.h>
#include <hip/hip_bf16.h>
#include <math.h>

#define IMGSZ   1024.0f
#define PRE     2000
#define POST    1000
#define NMS_THR 0.7f
#define MIN_SZ  16.0f
#define NBIN    1024
#define CAND    4096
#define NW      63          // 2000 bits -> 63 u32 words per row
#define NTILE   (PRE / 16)  // 125 column tiles

typedef float v2f __attribute__((ext_vector_type(2)));
typedef float v8f __attribute__((ext_vector_type(8)));

// ---------------------------------------------------------------- K0: zero per-call state
__global__ void zero_state_kernel(unsigned* __restrict__ hist, int* __restrict__ cnt, int nB) {
    int i = blockIdx.x * blockDim.x + threadIdx.x;
    int total = nB * NBIN;
    if (i < total) hist[i] = 0u;
    if (i < nB) cnt[i] = 0;
}

// ---------------------------------------------------------------- K1: decode (score-mask only)
__global__ __launch_bounds__(256)
void decode_score_kernel(const float4* __restrict__ anchors,
                         const float4* __restrict__ deltas,
                         const float*  __restrict__ scores,
                         float* __restrict__ mscore, int A, int total) {
    int gid = blockIdx.x * blockDim.x + threadIdx.x;
    if (gid >= total) return;
    int a = gid % A;
    float4 an = anchors[a];
    float4 dl = deltas[gid];
    float aw = an.z - an.x, ah = an.w - an.y;
    float ax = an.x + 0.5f * aw, ay = an.y + 0.5f * ah;
    float dw = fminf(dl.z, 4.0f), dh = fminf(dl.w, 4.0f);
    float px = dl.x * aw + ax, py = dl.y * ah + ay;
    float pw = __expf(dw) * aw, ph = __expf(dh) * ah;
    float x1 = fminf(fmaxf(px - 0.5f * pw, 0.0f), IMGSZ);
    float y1 = fminf(fmaxf(py - 0.5f * ph, 0.0f), IMGSZ);
    float x2 = fminf(fmaxf(px + 0.5f * pw, 0.0f), IMGSZ);
    float y2 = fminf(fmaxf(py + 0.5f * ph, 0.0f), IMGSZ);
    bool valid = (x2 - x1 >= MIN_SZ) && (y2 - y1 >= MIN_SZ);
    mscore[gid] = valid ? scores[gid] : -__builtin_inff();
}

// ---------------------------------------------------------------- K2a: per-image score histogram
__global__ __launch_bounds__(256)
void hist_kernel(const float* __restrict__ mscore, unsigned* __restrict__ hist, int A) {
    int img = blockIdx.y;
    __shared__ unsigned h[NBIN];
    for (int i = threadIdx.x; i < NBIN; i += 256) h[i] = 0u;
    __syncthreads();
    const float* ms = mscore + (size_t)img * A;
    int base = blockIdx.x * 2048;
    int end  = min(base + 2048, A);
    for (int i = base + (int)threadIdx.x; i < end; i += 256) {
        float s = ms[i];
        if (s >= 0.0f) {
            int bin = min((int)(s * (float)NBIN), NBIN - 1);
            atomicAdd(&h[bin], 1u);
        }
    }
    __syncthreads();
    for (int i = threadIdx.x; i < NBIN; i += 256)
        if (h[i]) atomicAdd(&hist[img * NBIN + i], h[i]);
}

// ---------------------------------------------------------------- K2b: find pre-NMS threshold bin
__global__ void thresh_kernel(const unsigned* __restrict__ hist, float* __restrict__ thrv) {
    int img = blockIdx.x;
    if (threadIdx.x == 0) {
        int s = 0, t = 0;
        for (int bin = NBIN - 1; bin >= 0; --bin) {
            s += (int)hist[img * NBIN + bin];
            if (s >= PRE) { t = bin; break; }
        }
        thrv[img] = (float)t / (float)NBIN;
    }
}

// ---------------------------------------------------------------- K2c: compact candidates
__global__ __launch_bounds__(256)
void compact_kernel(const float* __restrict__ mscore, const float* __restrict__ thrv,
                    float* __restrict__ cscore, int* __restrict__ cidx,
                    int* __restrict__ cnt, int A) {
    int img = blockIdx.y;
    int i = blockIdx.x * 256 + threadIdx.x;
    if (i >= A) return;
    float s = mscore[(size_t)img * A + i];
    if (s >= thrv[img]) {
        int pos = atomicAdd(&cnt[img], 1);
        if (pos < CAND) {
            cscore[img * CAND + pos] = s;
            cidx[img * CAND + pos]   = i;
        }
    }
}

// ---------------------------------------------------------------- K2d: bitonic sort + gather/decode top PRE
__global__ __launch_bounds__(1024)
void sort_gather_kernel(const float* __restrict__ cscore, const int* __restrict__ cidx,
                        const int* __restrict__ cnt,
                        const float4* __restrict__ anchors, const float4* __restrict__ deltas,
                        float* __restrict__ props, float* __restrict__ pscore,
                        float* __restrict__ parea, int A) {
    int img = blockIdx.x;
    __shared__ float key[CAND];
    __shared__ int   val[CAND];
    int n = min(cnt[img], CAND);
    for (int i = threadIdx.x; i < CAND; i += 1024) {
        key[i] = (i < n) ? cscore[img * CAND + i] : -__builtin_inff();
        val[i] = (i < n) ? cidx[img * CAND + i]   : -1;
    }
    __syncthreads();
    for (int k = 2; k <= CAND; k <<= 1) {
        for (int j = k >> 1; j > 0; j >>= 1) {
            for (int i = threadIdx.x; i < CAND; i += 1024) {
                int ixj = i ^ j;
                if (ixj > i) {
                    bool descSeg = ((i & k) == 0);
                    float a = key[i], c = key[ixj];
                    bool sw = descSeg ? (a < c) : (a > c);
                    if (sw) {
                        key[i] = c; key[ixj] = a;
                        int tv = val[i]; val[i] = val[ixj]; val[ixj] = tv;
                    }
                }
            }
            __syncthreads();
        }
    }
    // gather: re-decode + clip only the winning PRE boxes
    for (int i = threadIdx.x; i < PRE; i += 1024) {
        float s = key[i];
        int idx = val[i];
        float x1 = 0.f, y1 = 0.f, x2 = 0.f, y2 = 0.f;
        if (idx >= 0) {
            float4 an = anchors[idx];
            float4 dl = deltas[(size_t)img * A + idx];
            float aw = an.z - an.x, ah = an.w - an.y;
            float ax = an.x + 0.5f * aw, ay = an.y + 0.5f * ah;
            float dw = fminf(dl.z, 4.0f), dh = fminf(dl.w, 4.0f);
            float px = dl.x * aw + ax, py = dl.y * ah + ay;
            float pw = __expf(dw) * aw, ph = __expf(dh) * ah;
            x1 = fminf(fmaxf(px - 0.5f * pw, 0.0f), IMGSZ);
            y1 = fminf(fmaxf(py - 0.5f * ph, 0.0f), IMGSZ);
            x2 = fminf(fmaxf(px + 0.5f * pw, 0.0f), IMGSZ);
            y2 = fminf(fmaxf(py + 0.5f * ph, 0.0f), IMGSZ);
        } else {
            s = -__builtin_inff();
        }
        size_t o = (size_t)img * PRE + i;
        props[o * 4 + 0] = x1; props[o * 4 + 1] = y1;
        props[o * 4 + 2] = x2; props[o * 4 + 3] = y2;
        pscore[o] = s;
        parea[o]  = (x2 - x1) * (y2 - y1);
    }
}

// ---------------------------------------------------------------- K4: NMS suppression mask (WMMA tiles)
// One wave per 16-row tile. Pairwise area sums (area_i + area_j) computed as a
// rank-2 matmul on the matrix core: D = [area_a | 1] x [[1..1],[area_b]].
// Row boxes live in registers; ballots stay in registers (wave-uniform) and are
// resolved with a cndmask select tree; each 32-bit mask word is built across two
// 16-col tiles and stored straight to global (no LDS in this kernel).
__global__ __launch_bounds__(32)
void nms_mask_kernel(const float* __restrict__ props, const float* __restrict__ parea,
                     unsigned* __restrict__ mask) {
    const int img  = blockIdx.y;
    const int rt   = blockIdx.x;          // 0..124
    const int lane = threadIdx.x;         // 0..31
    const int row0 = rt * 16;
    const float* pb = props + (size_t)img * PRE * 4;
    const float* pa = parea + (size_t)img * PRE;

    const int n  = lane & 15;
    const int hi = lane >> 4;

    // hoist this tile's 8 row boxes (per half-wave) into registers
    float rx1[8], ry1[8], rx2[8], ry2[8];
#pragma unroll
    for (int r = 0; r < 8; ++r) {
        const float* bp = pb + (size_t)(row0 + r + 8 * hi) * 4;
        rx1[r] = bp[0]; ry1[r] = bp[1]; rx2[r] = bp[2]; ry2[r] = bp[3];
    }

    // A operand (16x4 f32): K=0 col = area_a, K=1 col = 1, K=2/3 = 0
    v2f Aop;
    Aop.x = (lane < 16) ? pa[row0 + lane] : 0.0f;
    Aop.y = (lane < 16) ? 1.0f : 0.0f;

    unsigned* mrow = mask + ((size_t)img * PRE + row0) * NW;
    const int rsel = lane & 7;
    const int hsh  = (lane >> 3) & 1;     // which 16-bit half of the ballot

    for (int w = 0; w < NW; ++w) {
        unsigned word = 0u;
#pragma unroll
        for (int half = 0; half < 2; ++half) {
            const int t = 2 * w + half;
            if (t < NTILE) {
                const int col0 = t * 16;
                if (t + 2 < NTILE)
                    __builtin_prefetch(pb + (size_t)(col0 + 32) * 4, 0, 0);  // global_prefetch_b8

                const float* cp = pb + (size_t)(col0 + n) * 4;
                float bx1 = cp[0], by1 = cp[1], bx2 = cp[2], by2 = cp[3];
                float ab  = pa[col0 + n];

                // B operand (4x16 f32): row K=0 = ones, row K=1 = area_b, K=2/3 = 0
                v2f Bop;
                Bop.x = (lane < 16) ? 1.0f : 0.0f;
                Bop.y = (lane < 16) ? ab   : 0.0f;

                v8f acc = {};
                acc = __builtin_amdgcn_wmma_f32_16x16x4_f32(
                          false, Aop, false, Bop, (short)0, acc, false, false);

                unsigned bms[8];
#pragma unroll
                for (int r = 0; r < 8; ++r) {
                    float ltx = fmaxf(rx1[r], bx1), lty = fmaxf(ry1[r], by1);
                    float rbx = fminf(rx2[r], bx2), rby = fminf(ry2[r], by2);
                    float iw = fmaxf(rbx - ltx, 0.0f), ih = fmaxf(rby - lty, 0.0f);
                    float inter = iw * ih;
                    float sums  = acc[r];                    // area_i + area_j from WMMA
                    int gi = row0 + r + 8 * hi, gj = col0 + n;
                    // inter/(sums-inter+eps) > thr  <=>  inter*(1+thr) > thr*(sums+eps)
                    bool sup = (inter * (1.0f + NMS_THR) > NMS_THR * (sums + 1e-6f)) && (gj > gi);
                    bms[r] = __builtin_amdgcn_ballot_w32(sup);
                }
                // ballots are wave-uniform: lane m (<16) selects bms[m&7] via cndmask tree
                unsigned bm = bms[0];
#pragma unroll
                for (int q = 1; q < 8; ++q) bm = (rsel == q) ? bms[q] : bm;
                unsigned bits = (bm >> (16 * hsh)) & 0xFFFFu;
                word |= bits << (half * 16);
            }
        }
        if (lane < 16)
            mrow[(size_t)lane * NW + w] = word;
    }
}

// ---------------------------------------------------------------- K5: serial greedy reduce (1 wave/image)
__global__ __launch_bounds__(32)
void nms_reduce_kernel(const unsigned* __restrict__ mask, unsigned* __restrict__ keep) {
    int img = blockIdx.x;
    int lane = threadIdx.x;
    __shared__ unsigned removed[NW];
    for (int w = lane; w < NW; w += 32) removed[w] = 0u;
    __syncthreads();
    const unsigned* mrow = mask + (size_t)img * PRE * NW;
    for (int i = 0; i < PRE; ++i) {
        unsigned rw = removed[i >> 5];                   // uniform read
        if (!((rw >> (i & 31)) & 1u)) {
            for (int w = lane; w < NW; w += 32)
                removed[w] |= mrow[(size_t)i * NW + w];
        }
        __syncthreads();                                 // single wave -> cheap
    }
    unsigned* kout = keep + (size_t)img * NW;
    for (int w = lane; w < NW; w += 32) {
        unsigned k = ~removed[w];
        if (w == NW - 1) k &= 0xFFFFu;                   // bits >= PRE invalid
        kout[w] = k;
    }
}

// ---------------------------------------------------------------- K6: compact survivors into output
__global__ __launch_bounds__(256)
void output_kernel(const float* __restrict__ props, const float* __restrict__ pscore,
                   const unsigned* __restrict__ keep, float* __restrict__ out) {
    int img = blockIdx.x, tid = threadIdx.x;
    __shared__ unsigned kw[NW];
    __shared__ int pref[NW];
    if (tid < NW) kw[tid] = keep[(size_t)img * NW + tid];
    __syncthreads();
    if (tid == 0) {
        int s = 0;
        for (int w = 0; w < NW; ++w) { pref[w] = s; s += __popc(kw[w]); }
    }
    float* o = out + (size_t)img * POST * 5;
    for (int i = tid; i < POST * 5; i += 256) o[i] = 0.0f;   // harness poisons d_out once
    __syncthreads();
    const float* pb = props  + (size_t)img * PRE * 4;
    const float* ps = pscore + (size_t)img * PRE;
    for (int i = tid; i < PRE; i += 256) {
        int w = i >> 5, b = i & 31;
        if ((kw[w] >> b) & 1u) {
            float s = ps[i];
            if (s > -1e30f) {
                unsigned lower = (b == 0) ? 0u : (kw[w] & (0xFFFFFFFFu >> (32 - b)));
                int rank = pref[w] + __popc(lower);
                if (rank < POST) {
                    o[rank * 5 + 0] = pb[i * 4 + 0];
                    o[rank * 5 + 1] = pb[i * 4 + 1];
                    o[rank * 5 + 2] = pb[i * 4 + 2];
                    o[rank * 5 + 3] = pb[i * 4 + 3];
                    o[rank * 5 + 4] = s;
                }
            }
        }
    }
}

// ---------------------------------------------------------------- host launcher
extern "C" void kernel_launch(void* const* d_in, const int* in_sizes, int n_in,
                              void* d_out, int out_size, void* d_ws, size_t ws_size,
                              hipStream_t stream) {
    const float4* anchors = (const float4*)d_in[0];
    const float4* deltas  = (const float4*)d_in[1];
    const float*  scores  = (const float*)d_in[2];
    float* out = (float*)d_out;

    const int A  = in_sizes[0] / 4;          // 261888
    const int nB = in_sizes[2] / A;          // 16
    const int total = nB * A;

    // workspace carve-out (256B aligned slices)
    char* p = (char*)d_ws;
    auto take = [&](size_t bytes) { char* r = p; p += (bytes + 255) & ~(size_t)255; return r; };
    float*    mscore = (float*)    take((size_t)total * 4);
    unsigned* hist   = (unsigned*) take((size_t)nB * NBIN * 4);
    float*    thrv   = (float*)    take((size_t)nB * 4);
    int*      cnt    = (int*)      take((size_t)nB * 4);
    float*    cscore = (float*)    take((size_t)nB * CAND * 4);
    int*      cidx   = (int*)      take((size_t)nB * CAND * 4);
    float*    props  = (float*)    take((size_t)nB * PRE * 4 * 4);
    float*    pscr   = (float*)    take((size_t)nB * PRE * 4);
    float*    parea  = (float*)    take((size_t)nB * PRE * 4);
    unsigned* mask   = (unsigned*) take((size_t)nB * PRE * NW * 4);
    unsigned* keep   = (unsigned*) take((size_t)nB * NW * 4);
    (void)ws_size;

    // K0: zero per-call state (graph-replay safe)
    {
        int n = nB * NBIN;
        zero_state_kernel<<<(n + 255) / 256, 256, 0, stream>>>(hist, cnt, nB);
    }
    // K1: decode -> masked scores
    decode_score_kernel<<<(total + 255) / 256, 256, 0, stream>>>(
        anchors, deltas, scores, mscore, A, total);
    // K2a: histogram
    hist_kernel<<<dim3((A + 2047) / 2048, nB), 256, 0, stream>>>(mscore, hist, A);
    // K2b: threshold
    thresh_kernel<<<nB, 32, 0, stream>>>(hist, thrv);
    // K2c: compact candidates
    compact_kernel<<<dim3((A + 255) / 256, nB), 256, 0, stream>>>(
        mscore, thrv, cscore, cidx, cnt, A);
    // K2d: sort + gather/decode top PRE
    sort_gather_kernel<<<nB, 1024, 0, stream>>>(
        cscore, cidx, cnt, anchors, deltas, props, pscr, parea, A);
    // K4: suppression mask via WMMA tiles
    nms_mask_kernel<<<dim3(NTILE, nB), 32, 0, stream>>>(props, parea, mask);
    // K5: greedy reduce
    nms_reduce_kernel<<<nB, 32, 0, stream>>>(mask, keep);
    // K6: compact into (B, POST, 5) output
    output_kernel<<<nB, 256, 0, stream>>>(props, pscr, keep, out);
    (void)out_size; (void)n_in;
}